// ReticulumLayer_67662914781946
// MI455X (gfx1250) — compile-verified
//
#include <hip/hip_runtime.h>
#include <math.h>

// ---------------- problem constants ----------------
#define BATCH   2
#define NTOKN   4096          // tokens per batch
#define DDIM    512           // feature dim
#define FDIM    257           // rfft bins
#define FP      264           // padded F (multiple of 8)
#define KP      (2*FP)        // 528: [Re | Im] packed row length
#define NTOK    (BATCH*NTOKN) // 8192
#define ALPHA_C 0.3f
#define INV_T   0.5f          // 1/TEMPERATURE
#define LN_EPS_C   1e-5f
#define NORM_EPS_C 1e-8f

// gram staging geometry
#define FTS     68            // LDS row stride (dwords) for gram tiles: 4 mod 64 -> bank-conflict free
#define GRAM_LDS_BYTES ((128+128+64+64)*FTS*4)   // Rn,In,Rm,Im = 104448 B
#define OFF_RN  0
#define OFF_IN  (128*FTS*4)
#define OFF_RM  (2*128*FTS*4)
#define OFF_IM  (2*128*FTS*4 + 64*FTS*4)

// mix staging geometry: 32 rows of x, 512 dwords + 8-dword pad after each 256 dwords
#define XSTRIDE 528
#define MIX_LDS_BYTES (32*XSTRIDE*4)             // 67584 B

typedef float v2f __attribute__((ext_vector_type(2)));
typedef float v8f __attribute__((ext_vector_type(8)));
typedef unsigned int u32x4 __attribute__((ext_vector_type(4)));
typedef int i32x4 __attribute__((ext_vector_type(4)));
typedef int i32x8 __attribute__((ext_vector_type(8)));

// CDNA5 f32 WMMA: D(16x16 f32) = A(16x4 f32) x B(4x16 f32) + C
__device__ __forceinline__ v8f wmma4(v2f a, v2f b, v8f c) {
  return __builtin_amdgcn_wmma_f32_16x16x4_f32(false, a, false, b, (short)0, c, false, false);
}

__device__ __forceinline__ float wave_sum(float v) {
#pragma unroll
  for (int o = 16; o > 0; o >>= 1) v += __shfl_xor(v, o, 32);
  return v;
}

// ---------------------------------------------------------------------------
// Tensor Data Mover: 2-D tile load Global -> LDS (D# groups per CDNA5 ISA ch.8)
//   tile_w elems (4B each) x tile_h rows, global row stride = stride_elems.
//   Optional LDS padding: pad_amt_code+1 dwords inserted every 2^(pad_int_code+1)
//   dwords stored (rows must be a multiple of the interval).
// ---------------------------------------------------------------------------
__device__ __forceinline__ void tdm_load_2d(unsigned lds_off, const void* gptr,
                                            unsigned tile_w, unsigned tile_h,
                                            unsigned stride_elems,
                                            unsigned pad_int_code,
                                            unsigned pad_amt_code,
                                            bool pad_en) {
  unsigned long long ga = (unsigned long long)gptr;
  u32x4 g0;
  g0[0] = 1u;                                          // count=1, user descriptor
  g0[1] = lds_off;                                     // LDS byte address
  g0[2] = (unsigned)(ga & 0xFFFFFFFFu);                // global_addr[31:0]  (bits 95:64)
  g0[3] = (unsigned)((ga >> 32) & 0x1FFFFFFu) | (2u << 30); // addr[56:32] | type=2
  i32x8 g1;
  g1[0] = (int)((2u << 16) |                           // data_size = 4B, wg_mask=0
                (pad_en ? (1u << 20) : 0u) |
                (pad_int_code << 22) | (pad_amt_code << 25));
  g1[1] = (int)((tile_w & 0xFFFFu) << 16);             // tensor_dim0[15:0] @ bits63:48
  g1[2] = (int)(((tile_w >> 16) & 0xFFFFu) | ((tile_h & 0xFFFFu) << 16)); // dim0 hi | dim1 lo
  g1[3] = (int)(((tile_h >> 16) & 0xFFFFu) | ((tile_w & 0xFFFFu) << 16)); // dim1 hi | tile_dim0
  g1[4] = (int)(tile_h & 0xFFFFu);                     // tile_dim1 (tile_dim2 = 0)
  g1[5] = (int)stride_elems;                           // tensor_dim0_stride[31:0]
  g1[6] = 0;                                           // stride hi | dim1_stride lo
  g1[7] = 0;
  i32x4 z4 = {0, 0, 0, 0};
#if __clang_major__ >= 23
  i32x8 z8 = {0, 0, 0, 0, 0, 0, 0, 0};
  __builtin_amdgcn_tensor_load_to_lds(g0, g1, z4, z4, z8, 0);
#else
  __builtin_amdgcn_tensor_load_to_lds(g0, g1, z4, z4, 0);
#endif
}

// ---------------------------------------------------------------------------
// 1) Fold rfft into the spectral projection (angles depend on f*d mod 512):
//    Mt[g][d]    =  sum_f W[g,f] cos(2pi f d/D)
//    Mt[FP+g][d] = -sum_f W[g,f] sin(2pi f d/D)
// ---------------------------------------------------------------------------
__global__ void k_basis(const float* __restrict__ W, float* __restrict__ Mt) {
  __shared__ float cosT[512];
  __shared__ float sinT[512];
  for (int i = threadIdx.x; i < 512; i += 256) {
    float ang = (6.283185307179586f / 512.0f) * (float)i;
    cosT[i] = __cosf(ang);
    sinT[i] = __sinf(ang);
  }
  __syncthreads();
  int idx = blockIdx.x * 256 + threadIdx.x;        // [0, KP*DDIM)
  int g = idx / DDIM, d = idx - g * DDIM;
  bool isIm = (g >= FP);
  int gr = isIm ? (g - FP) : g;
  float acc = 0.0f;
  if (gr < FDIM) {
    const float* wr = W + (size_t)gr * FDIM;
    const float* tab = isIm ? sinT : cosT;
    int id = 0;
    for (int f = 0; f < FDIM; ++f) {
      acc += wr[f] * tab[id];
      id = (id + d) & 511;
    }
    if (isIm) acc = -acc;
  }
  Mt[idx] = acc;
}

// ---------------------------------------------------------------------------
// 2) Spectral projection GEMM: Z(8192 x 528) = X(8192 x 512) @ Mt^T
// ---------------------------------------------------------------------------
__global__ void k_proj(const float* __restrict__ X, const float* __restrict__ Mt,
                       float* __restrict__ Z) {
  int wv = blockIdx.x * 8 + (threadIdx.x >> 5);   // 16896 waves = 512 * 33 tiles
  int lane = threadIdx.x & 31;
  int lm = lane & 15, hi = lane >> 4;
  int tn = wv / 33, tg = wv - tn * 33;
  int n0 = tn * 16, g0 = tg * 16;
  const float* arow = X  + (size_t)(n0 + lm) * DDIM;
  const float* brow = Mt + (size_t)(g0 + lm) * DDIM;
  v8f acc = {0.f,0.f,0.f,0.f,0.f,0.f,0.f,0.f};
  for (int k = 0; k < DDIM; k += 4) {
    v2f a = *(const v2f*)(arow + k + 2 * hi);
    v2f b = *(const v2f*)(brow + k + 2 * hi);
    acc = wmma4(a, b, acc);
  }
#pragma unroll
  for (int v = 0; v < 8; ++v) {
    int row = n0 + v + 8 * hi;
    Z[(size_t)row * KP + g0 + lm] = acc[v];
  }
}

// ---------------------------------------------------------------------------
// 3) Per-token normalization: Z[row] *= rsqrt(||Z[row]||^2 + eps)
// ---------------------------------------------------------------------------
__global__ void k_norm(float* __restrict__ Z) {
  int t = blockIdx.x * 8 + (threadIdx.x >> 5);
  int lane = threadIdx.x & 31;
  float* row = Z + (size_t)t * KP;
  float ss = 0.f;
  for (int c = lane; c < KP; c += 32) { float v = row[c]; ss += v * v; }
  ss = wave_sum(ss);
  float s = rsqrtf(ss + NORM_EPS_C);
  for (int c = lane; c < KP; c += 32) row[c] *= s;
}

// ---------------------------------------------------------------------------
// 4) Complex Gram + magnitude with TDM-staged LDS tiles.
//    Workgroup (8 waves, 4x2) computes 128x64 of S; wave tile 32x32.
//    Per f-chunk: waves 0..3 each TDM-load one tile (Rn/In/Rm/Im) into LDS,
//    s_wait_tensorcnt, barrier, then WMMA fragments come from LDS.
//    ReG = R R^T + I I^T ; ImG = I R^T - R I^T ; S = sqrt(Re^2+Im^2)
// ---------------------------------------------------------------------------
__global__ void k_gram(const float* __restrict__ Z, float* __restrict__ S) {
  extern __shared__ float smem[];
  float* Rn = smem;                  // 128 x FTS
  float* In = smem + 128 * FTS;      // 128 x FTS
  float* Rm = smem + 2 * 128 * FTS;  //  64 x FTS
  float* Im = Rm + 64 * FTS;         //  64 x FTS

  int blk = blockIdx.x;              // 4096 = 2 * 32 * 64
  int b = blk >> 11;
  int r = blk & 2047;
  int n0 = (r >> 6) * 128;
  int m0 = (r & 63) * 64;
  const float* Zb = Z + (size_t)b * NTOKN * KP;

  int w = threadIdx.x >> 5, lane = threadIdx.x & 31;
  int wi = w >> 1, wj = w & 1;       // 4 x 2 wave grid
  int lm = lane & 15, hi = lane >> 4;

  const float* rnp = Rn + (32 * wi + lm) * FTS + 2 * hi;
  const float* inp = In + (32 * wi + lm) * FTS + 2 * hi;
  const float* rmp = Rm + (32 * wj + lm) * FTS + 2 * hi;
  const float* imp = Im + (32 * wj + lm) * FTS + 2 * hi;

  v8f zero = {0.f,0.f,0.f,0.f,0.f,0.f,0.f,0.f};
  v8f aRe[2][2], aIm[2][2];
#pragma unroll
  for (int i = 0; i < 2; ++i)
#pragma unroll
    for (int j = 0; j < 2; ++j) { aRe[i][j] = zero; aIm[i][j] = zero; }

  for (int c = 0; c < 5; ++c) {      // f-chunks: 64,64,64,64,8  (covers FP=264)
    int f0 = c << 6;
    int ft = (c == 4) ? 8 : 64;
    unsigned icode = (c == 4) ? 2u : 5u;   // pad interval: 8 / 64 dwords
    unsigned acode = (c == 4) ? 59u : 3u;  // pad amount: 60 / 4 dwords -> stride 68
    __syncthreads();                 // previous chunk fully consumed
    if (w < 4) {
      const float* g; unsigned lds; unsigned h;
      if      (w == 0) { g = Zb + (size_t)n0 * KP + f0;      lds = OFF_RN; h = 128; }
      else if (w == 1) { g = Zb + (size_t)n0 * KP + FP + f0; lds = OFF_IN; h = 128; }
      else if (w == 2) { g = Zb + (size_t)m0 * KP + f0;      lds = OFF_RM; h = 64;  }
      else             { g = Zb + (size_t)m0 * KP + FP + f0; lds = OFF_IM; h = 64;  }
      tdm_load_2d(lds, g, ft, h, KP, icode, acode, true);
      __builtin_amdgcn_s_wait_tensorcnt(0);
    }
    __syncthreads();                 // tiles visible to all waves
    for (int ks = 0; ks < ft; ks += 4) {
      v2f Ra[2], Ia[2], Rb[2], Ib[2], nRa[2];
#pragma unroll
      for (int i = 0; i < 2; ++i) {
        Ra[i] = *(const v2f*)(rnp + 16 * i * FTS + ks);
        Ia[i] = *(const v2f*)(inp + 16 * i * FTS + ks);
        nRa[i] = -Ra[i];
        Rb[i] = *(const v2f*)(rmp + 16 * i * FTS + ks);
        Ib[i] = *(const v2f*)(imp + 16 * i * FTS + ks);
      }
#pragma unroll
      for (int i = 0; i < 2; ++i)
#pragma unroll
        for (int j = 0; j < 2; ++j) {
          aRe[i][j] = wmma4(Ra[i], Rb[j], aRe[i][j]);
          aRe[i][j] = wmma4(Ia[i], Ib[j], aRe[i][j]);
          aIm[i][j] = wmma4(Ia[i], Rb[j], aIm[i][j]);
          aIm[i][j] = wmma4(nRa[i], Ib[j], aIm[i][j]);
        }
    }
  }

  float* Sb = S + (size_t)b * NTOKN * NTOKN;
#pragma unroll
  for (int i = 0; i < 2; ++i)
#pragma unroll
    for (int j = 0; j < 2; ++j)
#pragma unroll
      for (int v = 0; v < 8; ++v) {
        int rr = n0 + 32 * wi + 16 * i + v + 8 * hi;
        int cc = m0 + 32 * wj + 16 * j + lm;
        float re = aRe[i][j][v], im = aIm[i][j][v];
        Sb[(size_t)rr * NTOKN + cc] = sqrtf(re * re + im * im);
      }
}

// ---------------------------------------------------------------------------
// 5) Softmax row statistics over S: mx[row], sm[row] = sum exp((S-mx)/T)
// ---------------------------------------------------------------------------
__global__ void k_rowstats(const float* __restrict__ S, float* __restrict__ mx,
                           float* __restrict__ sm) {
  __shared__ float red[256];
  int row = blockIdx.x;
  const float* sr = S + (size_t)row * NTOKN;
  int t = threadIdx.x;
  float m = -1e30f;
  for (int c = t; c < NTOKN; c += 256) m = fmaxf(m, sr[c]);
  red[t] = m; __syncthreads();
  for (int o = 128; o > 0; o >>= 1) {
    if (t < o) red[t] = fmaxf(red[t], red[t + o]);
    __syncthreads();
  }
  float mrow = red[0]; __syncthreads();
  float s = 0.f;
  for (int c = t; c < NTOKN; c += 256) s += __expf((sr[c] - mrow) * INV_T);
  red[t] = s; __syncthreads();
  for (int o = 128; o > 0; o >>= 1) {
    if (t < o) red[t] += red[t + o];
    __syncthreads();
  }
  if (t == 0) { mx[row] = mrow; sm[row] = red[0]; }
}

// ---------------------------------------------------------------------------
// 6) Mixing GEMM with on-the-fly softmax + residual blend.
//    h = (1-a)*x + a * softmax(S/T) @ x  -> out (pre-LayerNorm)
//    Workgroup: 32 rows x 512 cols; per 32-row K-chunk, wave0 TDM-loads
//    x[m0..m0+32) into LDS (512-dword rows, 8-dword pad each 256 -> stride 528).
// ---------------------------------------------------------------------------
__global__ void k_mix(const float* __restrict__ S, const float* __restrict__ X,
                      const float* __restrict__ mx, const float* __restrict__ sm,
                      float* __restrict__ out) {
  extern __shared__ float xs[];      // 32 x XSTRIDE
  int blk = blockIdx.x;              // 256 = 2 batches * 128 row-blocks
  int b = blk >> 7;
  int n0 = (blk & 127) * 32;
  int w = threadIdx.x >> 5, lane = threadIdx.x & 31;
  int wi = w >> 2, wj = w & 3;
  int lm = lane & 15, hi = lane >> 4;

  size_t grow = (size_t)b * NTOKN + (n0 + 16 * wi + lm);  // this lane's A row
  const float* srow = S + grow * NTOKN;
  float mv = mx[grow];
  float rs = 1.0f / sm[grow];
  const float* xb = X + (size_t)b * NTOKN * DDIM;

  int d0a[8];
#pragma unroll
  for (int j = 0; j < 8; ++j) {
    int d0 = 128 * wj + 16 * j + lm;
    d0a[j] = d0 + ((d0 >> 8) << 3);  // account for mid-row LDS pad
  }

  v8f zero = {0.f,0.f,0.f,0.f,0.f,0.f,0.f,0.f};
  v8f acc[8];
#pragma unroll
  for (int j = 0; j < 8; ++j) acc[j] = zero;

  for (int m0c = 0; m0c < NTOKN; m0c += 32) {
    __syncthreads();                 // previous x tile consumed
    if (w == 0) {
      tdm_load_2d(0u, xb + (size_t)m0c * DDIM, DDIM, 32, DDIM,
                  7u /*256 dw*/, 7u /*8 dw*/, true);
      __builtin_amdgcn_s_wait_tensorcnt(0);
    }
    __syncthreads();
#pragma unroll
    for (int mm = 0; mm < 32; mm += 4) {
      v2f s2 = *(const v2f*)(srow + m0c + mm + 2 * hi);
      v2f a;
      a.x = __expf((s2.x - mv) * INV_T) * rs;
      a.y = __expf((s2.y - mv) * INV_T) * rs;
      const float* xr = xs + (size_t)(mm + 2 * hi) * XSTRIDE;
#pragma unroll
      for (int j = 0; j < 8; ++j) {
        v2f bf;
        bf.x = xr[d0a[j]];
        bf.y = xr[d0a[j] + XSTRIDE];
        acc[j] = wmma4(a, bf, acc[j]);
      }
    }
  }

#pragma unroll
  for (int j = 0; j < 8; ++j)
#pragma unroll
    for (int v = 0; v < 8; ++v) {
      int rr = n0 + 16 * wi + v + 8 * hi;
      int dd = 128 * wj + 16 * j + lm;
      size_t off = ((size_t)b * NTOKN + rr) * DDIM + dd;
      float xv = X[off];
      out[off] = (1.0f - ALPHA_C) * xv + ALPHA_C * acc[j][v];
    }
}

// ---------------------------------------------------------------------------
// 7) In-place LayerNorm over D=512, one wave per token.
// ---------------------------------------------------------------------------
__global__ void k_ln(float* __restrict__ out, const float* __restrict__ gamma,
                     const float* __restrict__ beta) {
  int t = blockIdx.x * 8 + (threadIdx.x >> 5);
  int lane = threadIdx.x & 31;
  float* row = out + (size_t)t * DDIM;
  float h[16];
  float s = 0.f;
#pragma unroll
  for (int i = 0; i < 16; ++i) { h[i] = row[lane + 32 * i]; s += h[i]; }
  s = wave_sum(s);
  float mu = s * (1.0f / 512.0f);
  float q = 0.f;
#pragma unroll
  for (int i = 0; i < 16; ++i) { float d = h[i] - mu; q += d * d; }
  q = wave_sum(q);
  float rstd = rsqrtf(q * (1.0f / 512.0f) + LN_EPS_C);
#pragma unroll
  for (int i = 0; i < 16; ++i) {
    int c = lane + 32 * i;
    row[c] = (h[i] - mu) * rstd * gamma[c] + beta[c];
  }
}

// ---------------------------------------------------------------------------
extern "C" void kernel_launch(void* const* d_in, const int* in_sizes, int n_in,
                              void* d_out, int out_size, void* d_ws, size_t ws_size,
                              hipStream_t stream) {
  const float* x     = (const float*)d_in[0];  // (2,4096,512)
  const float* Wspec = (const float*)d_in[1];  // (257,257)
  const float* gamma = (const float*)d_in[2];  // (512,)
  const float* beta  = (const float*)d_in[3];  // (512,)

  float* out = (float*)d_out;                         // (2,4096,512)
  float* S   = out + (size_t)NTOK * DDIM;             // (2,4096,4096)

  float* ws = (float*)d_ws;
  float* Mt  = ws;                                    // KP x D      = 270336 f
  float* Z   = Mt + (size_t)KP * DDIM;                // NTOK x KP   = 4325376 f
  float* mxp = Z + (size_t)NTOK * KP;                 // NTOK
  float* smp = mxp + NTOK;                            // NTOK

  k_basis   <<<(KP * DDIM) / 256, 256, 0, stream>>>(Wspec, Mt);
  k_proj    <<<2112, 256, 0, stream>>>(x, Mt, Z);
  k_norm    <<<NTOK / 8, 256, 0, stream>>>(Z);
  k_gram    <<<4096, 256, GRAM_LDS_BYTES, stream>>>(Z, S);   // 2*32*64 wg tiles
  k_rowstats<<<NTOK, 256, 0, stream>>>(S, mxp, smp);
  k_mix     <<<BATCH * (NTOKN / 32), 256, MIX_LDS_BYTES, stream>>>(S, x, mxp, smp, out);
  k_ln      <<<NTOK / 8, 256, 0, stream>>>(out, gamma, beta);
}